// StnBilinearSampler_13065290514713
// MI455X (gfx1250) — compile-verified
//
#include <hip/hip_runtime.h>
#include <stdint.h>

typedef float f32x4 __attribute__((ext_vector_type(4)));
typedef float f32x2 __attribute__((ext_vector_type(2)));

#define TPB 256
#define PPT 64   // pixels per tile (4 lanes per pixel)

__global__ __launch_bounds__(TPB) void stn_bilinear_kernel(
    const float* __restrict__ images,    // [B,H,W,C] = [16,512,512,16]
    const float* __restrict__ sampling,  // [B,Ho,Wo,2] = [16,384,384,2]
    float* __restrict__ out,             // [B,Ho,Wo,C]
    int nTiles)
{
    constexpr int H = 512, W = 512;
    constexpr int HoWo = 384 * 384;

    __shared__ f32x2 sbuf[2][PPT];   // double-buffered sampling coords

    const int tid = threadIdx.x;
    const int cg  = tid & 3;         // channel group (float4) within pixel
    const int pit = tid >> 2;        // pixel index within tile

    int tile = blockIdx.x;
    if (tile >= nTiles) return;

    // Prologue: async-stage tile0's coords into LDS buffer 0.
    // Producer lanes are tid<64 == waves 0 and 1 entirely (wave32).
    if (tid < PPT) {
        const f32x2* g = (const f32x2*)sampling + ((size_t)tile * PPT + tid);
        uint32_t lds = (uint32_t)(uintptr_t)&sbuf[0][tid];   // addr[31:0] == LDS byte offset
        asm volatile("global_load_async_to_lds_b64 %0, %1, off"
                     :: "v"(lds), "v"(g) : "memory");
    }

    int buf = 0;
    for (; tile < nTiles; tile += (int)gridDim.x) {
        const int nextTile = tile + (int)gridDim.x;
        const bool hasNext = nextTile < nTiles;

        if (tid < PPT) {
            if (hasNext) {
                // Prefetch next tile's coords into the other buffer, then
                // wait until the *current* tile's async load has landed
                // (allow the just-issued one to remain outstanding).
                const f32x2* g = (const f32x2*)sampling + ((size_t)nextTile * PPT + tid);
                uint32_t lds = (uint32_t)(uintptr_t)&sbuf[buf ^ 1][tid];
                asm volatile("global_load_async_to_lds_b64 %0, %1, off"
                             :: "v"(lds), "v"(g) : "memory");
                asm volatile("s_wait_asynccnt 0x1" ::: "memory");
            } else {
                asm volatile("s_wait_asynccnt 0x0" ::: "memory");
            }
        }
        __syncthreads();   // publish LDS coords to consumer waves

        const f32x2 s = sbuf[buf][pit];

        const int pid = tile * PPT + pit;     // global pixel id (< 2^22)
        const int b   = pid / HoWo;           // batch index

        // [-1,1] -> pixel coords
        const float xs = 0.5f * (s.x + 1.0f) * (float)(W - 1);
        const float ys = 0.5f * (s.y + 1.0f) * (float)(H - 1);

        const float x0f = floorf(xs);
        const float y0f = floorf(ys);
        const float fx = xs - x0f;   // weight toward x1  (== xs - x0f)
        const float fy = ys - y0f;   // weight toward y1
        const float gx = 1.0f - fx;  // == x1f - xs
        const float gy = 1.0f - fy;  // == y1f - ys

        int x0 = (int)x0f, y0 = (int)y0f;
        int x1 = x0 + 1,   y1 = y0 + 1;
        x0 = min(max(x0, 0), W - 1);
        x1 = min(max(x1, 0), W - 1);
        y0 = min(max(y0, 0), H - 1);
        y1 = min(max(y1, 0), H - 1);

        const float w_nw = gx * gy;
        const float w_ne = fx * gy;
        const float w_sw = gx * fy;
        const float w_se = fx * fy;

        // images as float4 units; per-pixel record is C=16 floats = 4 float4s.
        const f32x4* base = (const f32x4*)images + (size_t)b * (size_t)(H * W * 4) + cg;
        const f32x4 i_nw = base[(size_t)(y0 * W + x0) * 4];
        const f32x4 i_ne = base[(size_t)(y0 * W + x1) * 4];
        const f32x4 i_sw = base[(size_t)(y1 * W + x0) * 4];
        const f32x4 i_se = base[(size_t)(y1 * W + x1) * 4];

        f32x4 r;
        r.x = __builtin_fmaf(w_nw, i_nw.x, __builtin_fmaf(w_ne, i_ne.x,
              __builtin_fmaf(w_sw, i_sw.x, w_se * i_se.x)));
        r.y = __builtin_fmaf(w_nw, i_nw.y, __builtin_fmaf(w_ne, i_ne.y,
              __builtin_fmaf(w_sw, i_sw.y, w_se * i_se.y)));
        r.z = __builtin_fmaf(w_nw, i_nw.z, __builtin_fmaf(w_ne, i_ne.z,
              __builtin_fmaf(w_sw, i_sw.z, w_se * i_se.z)));
        r.w = __builtin_fmaf(w_nw, i_nw.w, __builtin_fmaf(w_ne, i_ne.w,
              __builtin_fmaf(w_sw, i_sw.w, w_se * i_se.w)));

        // Streaming output: non-temporal so the 151 MB write stream does not
        // evict the L2-resident 268 MB image set.
        f32x4* op = (f32x4*)out + ((size_t)pid * 4 + cg);
        __builtin_nontemporal_store(r, op);

        __syncthreads();   // all consumers done with sbuf[buf] before it is reused
        buf ^= 1;
    }
}

extern "C" void kernel_launch(void* const* d_in, const int* in_sizes, int n_in,
                              void* d_out, int out_size, void* d_ws, size_t ws_size,
                              hipStream_t stream) {
    (void)in_sizes; (void)n_in; (void)d_ws; (void)ws_size; (void)out_size;
    const float* images   = (const float*)d_in[0];
    const float* sampling = (const float*)d_in[1];
    float* out            = (float*)d_out;

    const int pixTotal = 16 * 384 * 384;        // 2,359,296
    const int nTiles   = pixTotal / PPT;        // 36,864 (exact)
    const int grid     = 4608;                  // 8 tiles per persistent block

    stn_bilinear_kernel<<<grid, TPB, 0, stream>>>(images, sampling, out, nTiles);
}